// NetworkBasic_90348932038962
// MI455X (gfx1250) — compile-verified
//
#include <hip/hip_runtime.h>

typedef __attribute__((ext_vector_type(2))) float v2f;
typedef __attribute__((ext_vector_type(8))) float v8f;

namespace {

constexpr int NB = 2, C = 8, H = 128, W = 128, T = 64;
constexpr int HW   = H * W;        // 16384
constexpr int CHW  = C * HW;       // 131072
constexpr int NCHW = NB * CHW;     // 262144

// SRM / LIF constants (Ts = 1)
constexpr float kE    = 2.71828182845904523536f;
constexpr float kD1   = 0.36787944117144233f;   // exp(-Ts/1)
constexpr float kD2   = 0.60653065971263342f;   // exp(-Ts/2)
constexpr float kC1   = kE;                     // e*Ts/tau*Ts, tau=1
constexpr float kC2   = kE * 0.5f;              // tau=2
constexpr float kRef1 = 30.0f * kE;             // scaleRef*theta*e*Ts/tauRef (theta=30, tau=1)
constexpr float kRef2 = 25.0f * kE;             // theta=50, tau=2 -> 50*e/2
constexpr float kTh1  = 30.0f;
constexpr float kTh2  = 50.0f;

} // namespace

// ---------------------------------------------------------------------------
// Stage 1: psp1 = alpha_filter(x, tau=1).
// in:  [N,C,H,W,T] (T contiguous) ; out: [T][NCHW] plane-major.
// ---------------------------------------------------------------------------
__global__ __launch_bounds__(256) void psp1_kernel(const float* __restrict__ in,
                                                   float* __restrict__ psp)
{
    int pix = blockIdx.x * 256 + threadIdx.x;
    const float4* in4 = (const float4*)(in + (size_t)pix * T);
    float P = 0.f, Q = 0.f;
    for (int tb = 0; tb < T / 4; ++tb) {
        float4 x = in4[tb];
        float xs[4] = {x.x, x.y, x.z, x.w};
#pragma unroll
        for (int j = 0; j < 4; ++j) {
            Q = kD1 * (Q + P);          // uses P before injecting x[t] (zero at lag 0)
            P = kD1 * P + xs[j];
            psp[(tb * 4 + j) * NCHW + pix] = kC1 * Q;
        }
    }
}

// ---------------------------------------------------------------------------
// Conv2d (KSxKS, pad, 8->8 ch) per timestep via implicit GEMM on
// V_WMMA_F32_16X16X4_F32.  M = 16 pixels of one row, N = 16 out-channels
// (8 real + 8 zero-pad), K = KS*KS*8 chunked by 4 (ci fastest within a tap).
// Block = 256 threads = 8 waves; tile = 8 rows x 16 cols per (t,n).
// ---------------------------------------------------------------------------
template <int KS, int PAD>
__global__ __launch_bounds__(256) void conv_wmma_kernel(const float* __restrict__ src,
                                                        const float* __restrict__ w,
                                                        float* __restrict__ dst)
{
    constexpr int TR = 8 + 2 * PAD;    // tile rows incl. halo
    constexpr int TC = 16 + 2 * PAD;   // tile cols incl. halo
    constexpr int P2 = KS * KS;        // taps

    __shared__ float tile[TR][TC][C];      // [row][col][ci]
    __shared__ float wl[P2][C][16];        // [tap][ci][co], co 8..15 zero

    const int tid = threadIdx.x;
    const int x0  = blockIdx.x * 16;
    const int y0  = blockIdx.y * 8;
    const int t   = blockIdx.z >> 1;
    const int n   = blockIdx.z & 1;
    const int planeBase = t * NCHW + n * CHW;

    // Stage weights -> LDS, re-laid out [p][ci][co] with zero-padded co.
    for (int i = tid; i < P2 * C * 16; i += 256) {
        int co = i & 15;
        int ci = (i >> 4) & 7;
        int p  = i >> 7;
        wl[p][ci][co] = (co < C) ? w[(co * C + ci) * P2 + p] : 0.f;
    }
    // Stage input tile (+halo, zero padded at image border) -> LDS.
    for (int i = tid; i < C * TR * TC; i += 256) {
        int ci = i / (TR * TC);
        int r  = i % (TR * TC);
        int ry = r / TC, rx = r % TC;
        int gy = y0 + ry - PAD, gx = x0 + rx - PAD;
        float v = 0.f;
        if (gy >= 0 && gy < H && gx >= 0 && gx < W)
            v = src[planeBase + ci * HW + gy * W + gx];
        tile[ry][rx][ci] = v;
    }
    __syncthreads();

    const int wave = tid >> 5;          // output row within tile
    const int lane = tid & 31;
    const int ln   = lane & 15;         // A: pixel m ; B: out-channel n
    const int half = lane >> 4;         // K-half select (K0/K1 vs K2/K3)

    v8f acc = {};
    for (int p = 0; p < P2; ++p) {
        const int kh = p / KS, kw = p % KS;
        const float* trow = &tile[wave + kh][ln + kw][0];
#pragma unroll
        for (int cb = 0; cb < C; cb += 4) {
            const int ci = cb + 2 * half;
            v2f a, b;
            a.x = trow[ci];             // A[m][k]   k = p*8 + ci
            a.y = trow[ci + 1];
            b.x = wl[p][ci][ln];        // B[k][co]
            b.y = wl[p][ci + 1][ln];
            acc = __builtin_amdgcn_wmma_f32_16x16x4_f32(
                false, a, false, b, (short)0, acc, false, false);
        }
    }

    // D layout: lane co = ln, rows m = vg + 8*half.
    if (ln < C) {
        const int y = y0 + wave;
        const int base = planeBase + ln * HW + y * W + x0 + 8 * half;
#pragma unroll
        for (int vg = 0; vg < 8; ++vg) dst[base + vg] = acc[vg];
    }
}

// ---------------------------------------------------------------------------
// Stage 3: LIF spike (theta=30, tauRef=1) fused with psp2 alpha filter (tau=2).
// ---------------------------------------------------------------------------
__global__ __launch_bounds__(256) void lif1_psp2_kernel(const float* __restrict__ u,
                                                        float* __restrict__ psp)
{
    int pix = blockIdx.x * 256 + threadIdx.x;
    float Pr = 0.f, Qr = 0.f, P2 = 0.f, Q2 = 0.f;
    for (int t = 0; t < T; ++t) {
        float ut = u[t * NCHW + pix];
        Qr = kD1 * (Qr + Pr);                              // refractory kernel
        float s = (ut - kRef1 * Qr >= kTh1) ? 1.f : 0.f;   // Heaviside spike
        Pr = kD1 * Pr + s;
        Q2 = kD2 * (Q2 + P2);                              // psp2 alpha state
        P2 = kD2 * P2 + s;
        psp[t * NCHW + pix] = kC2 * Q2;
    }
}

// ---------------------------------------------------------------------------
// Stage 5: LIF spike (theta=50, tauRef=2) -> output [N,C,H,W,T].
// ---------------------------------------------------------------------------
__global__ __launch_bounds__(256) void lif2_out_kernel(const float* __restrict__ u,
                                                       float* __restrict__ out)
{
    int pix = blockIdx.x * 256 + threadIdx.x;
    float Pr = 0.f, Qr = 0.f;
    float sb[T];
#pragma unroll
    for (int t = 0; t < T; ++t) {
        float ut = u[t * NCHW + pix];
        Qr = kD2 * (Qr + Pr);
        float s = (ut - kRef2 * Qr >= kTh2) ? 1.f : 0.f;
        Pr = kD2 * Pr + s;
        sb[t] = s;
    }
    float4* o4 = (float4*)(out + (size_t)pix * T);
#pragma unroll
    for (int q = 0; q < T / 4; ++q)
        o4[q] = make_float4(sb[4 * q], sb[4 * q + 1], sb[4 * q + 2], sb[4 * q + 3]);
}

// ---------------------------------------------------------------------------
// Pipeline: psp1 -> conv1(WMMA) -> lif1+psp2 -> conv2(WMMA) -> lif2.
// Workspace: two [T][NCHW] f32 ping-pong buffers (2 x 64 MiB).
// ---------------------------------------------------------------------------
extern "C" void kernel_launch(void* const* d_in, const int* in_sizes, int n_in,
                              void* d_out, int out_size, void* d_ws, size_t ws_size,
                              hipStream_t stream)
{
    (void)in_sizes; (void)n_in; (void)out_size; (void)ws_size;

    const float* spikeIn = (const float*)d_in[0];   // [2,8,128,128,64]
    const float* w1      = (const float*)d_in[1];   // [8,8,5,5]
    const float* w2      = (const float*)d_in[2];   // [8,8,3,3]
    float* out  = (float*)d_out;
    float* bufA = (float*)d_ws;
    float* bufB = bufA + (size_t)T * NCHW;          // +16,777,216 floats

    const dim3 cgrid(W / 16, H / 8, T * NB);        // (8, 16, 128)

    psp1_kernel<<<NCHW / 256, 256, 0, stream>>>(spikeIn, bufA);
    conv_wmma_kernel<5, 2><<<cgrid, 256, 0, stream>>>(bufA, w1, bufB);
    lif1_psp2_kernel<<<NCHW / 256, 256, 0, stream>>>(bufB, bufA);
    conv_wmma_kernel<3, 1><<<cgrid, 256, 0, stream>>>(bufA, w2, bufB);
    lif2_out_kernel<<<NCHW / 256, 256, 0, stream>>>(bufB, out);
}